// GCNEncoder_12378095747244
// MI455X (gfx1250) — compile-verified
//
#include <hip/hip_runtime.h>
#include <hip/hip_bf16.h>
#include <stdint.h>

// ---------------------------------------------------------------------------
// GCN 2-layer forward for MI455X (gfx1250, wave32).
// Bandwidth-bound on edge gather/scatter (~3.4 GB @ 23.3 TB/s ≈ 150us floor);
// GEMMs (~5 GFLOP) run on V_WMMA_F32_16X16X4_F32 to keep f32 precision.
// ---------------------------------------------------------------------------

typedef float v2f __attribute__((ext_vector_type(2)));
typedef float v8f __attribute__((ext_vector_type(8)));

#define NNODES 50000
#define CH     128   // HID

// ---- degree / dinv ---------------------------------------------------------

__global__ __launch_bounds__(256) void k_fill_one(float* __restrict__ p, int n) {
    int i = blockIdx.x * 256 + threadIdx.x;
    if (i < n) p[i] = 1.0f;   // self-loop contributes 1 to every in-degree
}

__global__ __launch_bounds__(256) void k_deg_accum(const long long* __restrict__ col,
                                                   float* __restrict__ deg, int E) {
    int e = blockIdx.x * 256 + threadIdx.x;
    if (e < E) atomicAdd(&deg[(int)col[e]], 1.0f);
}

__global__ __launch_bounds__(256) void k_dinv(const float* __restrict__ deg,
                                              float* __restrict__ dinv, int n) {
    int i = blockIdx.x * 256 + threadIdx.x;
    if (i < n) dinv[i] = rsqrtf(deg[i]);   // deg >= 1 always (self-loops)
}

// ---- dense GEMM via V_WMMA_F32_16X16X4_F32 --------------------------------
// C[M,128] = A[M,K] @ B[K,128].  One wave computes a 16x16 tile.
// Block = 256 threads = 8 waves = the 8 column blocks; grid.x = M/16.
// K is a template constant (256 / 128) so the k-loop has no remainder path.
// A operand layout (16x4 f32): lanes 0-15 hold K=0,1; lanes 16-31 K=2,3.
// B operand layout (4x16 f32): vgpr0 = row K=0|2 across lanes, vgpr1 = K=1|3.
// C/D layout (16x16 f32): vgpr v, lanes 0-15 -> row v, lanes 16-31 -> row v+8.

template <int K>
__global__ __launch_bounds__(256) void k_gemm_wmma_f32(const float* __restrict__ A,
                                                       const float* __restrict__ B,
                                                       float* __restrict__ C) {
    const int lane = threadIdx.x & 31;
    const int wave = threadIdx.x >> 5;          // 0..7 : column block
    const int row0 = blockIdx.x << 4;
    const int col0 = wave << 4;
    const int lr   = lane & 15;
    const int hi   = lane >> 4;                  // 0 or 1

    const float* __restrict__ arow = A + (size_t)(row0 + lr) * (size_t)K;
    const float* __restrict__ bcol = B + col0 + lr;     // B is K x 128

    v8f acc = {0.f, 0.f, 0.f, 0.f, 0.f, 0.f, 0.f, 0.f};

    #pragma unroll 4
    for (int k = 0; k < K; k += 4) {
        const int ka = k + (hi << 1);            // even -> 8B aligned
        v2f a = *(const v2f*)(arow + ka);        // A[m][ka], A[m][ka+1]
        v2f b;
        b.x = bcol[(size_t)ka * CH];             // B[ka  ][col]
        b.y = bcol[(size_t)(ka + 1) * CH];       // B[ka+1][col]
        acc = __builtin_amdgcn_wmma_f32_16x16x4_f32(
            /*neg_a=*/false, a, /*neg_b=*/false, b,
            /*c_mod=*/(short)0, acc, /*reuse_a=*/false, /*reuse_b=*/false);
    }

    float* __restrict__ crow = C + (size_t)(row0 + (hi << 3)) * CH + col0 + lr;
    #pragma unroll
    for (int v = 0; v < 8; ++v)
        crow[(size_t)v * CH] = acc[v];
}

// ---- aggregation -----------------------------------------------------------

// out[i][c] = H[i][c] * dinv[i]^2 + bias[c]   (self-loop term + bias)
__global__ __launch_bounds__(256) void k_agg_init(const float* __restrict__ H,
                                                  const float* __restrict__ dinv,
                                                  const float* __restrict__ bias,
                                                  float* __restrict__ out, int total) {
    int idx = blockIdx.x * 256 + threadIdx.x;
    if (idx >= total) return;
    int node = idx >> 7;
    int c    = idx & (CH - 1);
    float d = dinv[node];
    out[idx] = H[idx] * d * d + bias[c];
}

// One wave (32 lanes) per edge; lane handles 4 channels: float4 gather + 4
// atomics.  8 edges per 256-thread block.  This is the roofline-critical
// kernel: ~845 MB gather + ~845 MB scatter-atomics per layer.
__global__ __launch_bounds__(256) void k_agg_edges(const float* __restrict__ H,
                                                   const float* __restrict__ dinv,
                                                   const long long* __restrict__ row,
                                                   const long long* __restrict__ col,
                                                   float* __restrict__ out, int E) {
    int e = blockIdx.x * 8 + (threadIdx.x >> 5);
    if (e >= E) return;
    int lane = threadIdx.x & 31;
    int r = (int)row[e];
    int c = (int)col[e];
    float nrm = dinv[r] * dinv[c];
    const float4 h4 = *(const float4*)(H + (size_t)r * CH + lane * 4);
    float* o = out + (size_t)c * CH + lane * 4;
    atomicAdd(o + 0, h4.x * nrm);
    atomicAdd(o + 1, h4.y * nrm);
    atomicAdd(o + 2, h4.z * nrm);
    atomicAdd(o + 3, h4.w * nrm);
}

__global__ __launch_bounds__(256) void k_relu(float* __restrict__ p, int n) {
    int i = blockIdx.x * 256 + threadIdx.x;
    if (i < n) p[i] = fmaxf(p[i], 0.0f);
}

// ---------------------------------------------------------------------------

extern "C" void kernel_launch(void* const* d_in, const int* in_sizes, int n_in,
                              void* d_out, int out_size, void* d_ws, size_t ws_size,
                              hipStream_t stream) {
    const float*     x    = (const float*)d_in[0];
    const long long* eidx = (const long long*)d_in[1];   // int64, shape (2, E)
    const float*     W1   = (const float*)d_in[2];
    const float*     b1   = (const float*)d_in[3];
    const float*     W2   = (const float*)d_in[4];
    const float*     b2   = (const float*)d_in[5];
    float*           out  = (float*)d_out;

    const int N = NNODES;
    const int E = in_sizes[1] / 2;

    const long long* row = eidx;             // sources
    const long long* col = eidx + E;         // targets

    float* deg  = (float*)d_ws;              // [N]
    float* dinv = deg + N;                   // [N]
    float* H    = dinv + N;                  // [N*128]  GEMM output (both layers)
    float* HA   = H + (size_t)N * CH;        // [N*128]  layer-1 aggregated+relu

    const int nc = N * CH;                   // 6.4M elements

    // degree -> dinv
    k_fill_one <<<(N + 255) / 256, 256, 0, stream>>>(deg, N);
    k_deg_accum<<<(E + 255) / 256, 256, 0, stream>>>(col, deg, E);
    k_dinv     <<<(N + 255) / 256, 256, 0, stream>>>(deg, dinv, N);

    // ---- layer 1: H = X @ W1 ; HA = norm-agg(H) + b1 ; relu ----
    k_gemm_wmma_f32<256><<<N / 16, 256, 0, stream>>>(x, W1, H);
    k_agg_init <<<(nc + 255) / 256, 256, 0, stream>>>(H, dinv, b1, HA, nc);
    k_agg_edges<<<(E + 7) / 8, 256, 0, stream>>>(H, dinv, row, col, HA, E);
    k_relu     <<<(nc + 255) / 256, 256, 0, stream>>>(HA, nc);

    // ---- layer 2: H = HA @ W2 ; out = norm-agg(H) + b2 ----
    k_gemm_wmma_f32<128><<<N / 16, 256, 0, stream>>>(HA, W2, H);
    k_agg_init <<<(nc + 255) / 256, 256, 0, stream>>>(H, dinv, b2, out, nc);
    k_agg_edges<<<(E + 7) / 8, 256, 0, stream>>>(H, dinv, row, col, out, E);
}